// SmoothMaxPool3D_59605556134310
// MI455X (gfx1250) — compile-verified
//
#include <hip/hip_runtime.h>
#include <hip/hip_bf16.h>

// SmoothMaxPool3D (logsumexp pool), kernel=2, stride=2, T=0.01
// x: [8,32,64,64,64] f32 -> out: [8,32,32,32,32] f32
// Pure streaming op (each input read exactly once): staged through LDS with
// CDNA5 async loads (ASYNCcnt path), computed with stable logsumexp.

typedef float v2f __attribute__((ext_vector_type(2)));
typedef float v4f __attribute__((ext_vector_type(4)));

__device__ __forceinline__ float smax8(float a0, float a1, float b0, float b1,
                                       float c0, float c1, float d0, float d1) {
    // m = max of 8 taps (compiler forms v_max3_f32 chains)
    float m = fmaxf(fmaxf(fmaxf(a0, a1), fmaxf(b0, b1)),
                    fmaxf(fmaxf(c0, c1), fmaxf(d0, d1)));
    const float invT = 100.0f;
    float s = __expf((a0 - m) * invT) + __expf((a1 - m) * invT) +
              __expf((b0 - m) * invT) + __expf((b1 - m) * invT) +
              __expf((c0 - m) * invT) + __expf((c1 - m) * invT) +
              __expf((d0 - m) * invT) + __expf((d1 - m) * invT);
    return fmaf(0.01f, __logf(s), m);
}

__global__ __launch_bounds__(256) void smoothmaxpool3d_async_kernel(
    const float* __restrict__ x, float* __restrict__ out) {
    // 16 KB staging: 4 x 16B slots per thread, lane-private (no barrier needed).
    __shared__ v4f buf[256 * 4];

    const unsigned tid = blockIdx.x * 256u + threadIdx.x;
    // tid -> (bc, oz, oy, px): 256 * 32 * 32 * 16 = 4,194,304 threads
    const unsigned px = tid & 15u;          // output x-pair: ox = 2*px, 2*px+1
    unsigned r = tid >> 4;
    const unsigned oy = r & 31u; r >>= 5;
    const unsigned oz = r & 31u; r >>= 5;
    const unsigned bc = r;                  // fused batch*channel in [0,256)

    // input element offset of (bc, iz=2*oz, iy=2*oy, ix=4*px); strides W=64, H*W=4096
    const unsigned e0  = ((bc * 64u + 2u * oz) * 64u + 2u * oy) * 64u + 4u * px;
    const unsigned b00 = e0 * 4u;           // byte offsets for the 4 (dz,dy) rows
    const unsigned b01 = b00 + 256u;        // +1 row   (64 floats)
    const unsigned b10 = b00 + 16384u;      // +1 slice (4096 floats)
    const unsigned b11 = b10 + 256u;

    const unsigned slot = threadIdx.x * 4u;
    // Flat LDS pointer truncated to 32 bits == wave-relative LDS byte offset.
    const unsigned l0 = (unsigned)(uintptr_t)&buf[slot];

    // CDNA5 async global->LDS staging (GVS mode: SGPR base + 32-bit VGPR offset).
    asm volatile(
        "global_load_async_to_lds_b128 %0, %4, %8\n\t"
        "global_load_async_to_lds_b128 %1, %5, %8\n\t"
        "global_load_async_to_lds_b128 %2, %6, %8\n\t"
        "global_load_async_to_lds_b128 %3, %7, %8"
        :
        : "v"(l0), "v"(l0 + 16u), "v"(l0 + 32u), "v"(l0 + 48u),
          "v"(b00), "v"(b01), "v"(b10), "v"(b11),
          "s"(x)
        : "memory");

    asm volatile("s_wait_asynccnt 0" ::: "memory");

    const v4f q00 = buf[slot + 0];   // iz=2z,   iy=2y,   ix=4p..4p+3
    const v4f q01 = buf[slot + 1];   // iz=2z,   iy=2y+1
    const v4f q10 = buf[slot + 2];   // iz=2z+1, iy=2y
    const v4f q11 = buf[slot + 3];   // iz=2z+1, iy=2y+1

    const float o0 = smax8(q00.x, q00.y, q01.x, q01.y, q10.x, q10.y, q11.x, q11.y);
    const float o1 = smax8(q00.z, q00.w, q01.z, q01.w, q10.z, q10.w, q11.z, q11.w);

    const unsigned oidx = ((bc * 32u + oz) * 32u + oy) * 32u + 2u * px;
    v2f ov;
    ov.x = o0;
    ov.y = o1;
    __builtin_nontemporal_store(ov, (v2f*)(out + oidx));   // streaming NT store
}

extern "C" void kernel_launch(void* const* d_in, const int* in_sizes, int n_in,
                              void* d_out, int out_size, void* d_ws, size_t ws_size,
                              hipStream_t stream) {
    (void)in_sizes; (void)n_in; (void)out_size; (void)d_ws; (void)ws_size;
    const float* x = (const float*)d_in[0];
    float* out = (float*)d_out;
    // 4,194,304 threads total: 16384 blocks x 256 threads, 2 outputs/thread.
    smoothmaxpool3d_async_kernel<<<dim3(16384), dim3(256), 0, stream>>>(x, out);
}